// Evaluate_9225589752230
// MI455X (gfx1250) — compile-verified
//
#include <hip/hip_runtime.h>
#include <hip/hip_bf16.h>

#define NN 16384
#define DD 512
#define TK 10
#define KC 32
#define ASTR 36              // 32 + 4 pad floats: conflict-free ds_load_b64 fragments
#define NCHUNK (DD / KC)     // 16

typedef __attribute__((ext_vector_type(2))) float v2f;
typedef __attribute__((ext_vector_type(8))) float v8f;

#if defined(__has_builtin)
#if __has_builtin(__builtin_amdgcn_global_load_async_to_lds_b128)
#define HAVE_ASYNC_LDS 1
#endif
#endif

#ifdef HAVE_ASYNC_LDS
typedef int v4i __attribute__((vector_size(16)));
typedef __attribute__((address_space(1))) v4i* gv4p;   // global (AS1) int4*
typedef __attribute__((address_space(3))) v4i* lv4p;   // LDS (AS3) int4*
#endif

// ---------------------------------------------------------------------------
// Kernel 1: sim = L @ R^T via V_WMMA_F32_16X16X4_F32.
// 256 threads (8 waves), block tile 128x128, wave tile 64x32 (4x2 WMMA tiles).
// Double-buffered LDS; staging via GLOBAL_LOAD_ASYNC_TO_LDS_B128 when available.
// ---------------------------------------------------------------------------
__device__ __forceinline__ void compute_chunk(const float* __restrict__ Abuf,
                                              const float* __restrict__ Bbuf,
                                              const int (&aBase)[4],
                                              const int (&bBase)[2],
                                              v8f (&acc)[4][2]) {
    #pragma unroll
    for (int kk = 0; kk < KC; kk += 4) {
        v2f a[4], b[2];
        #pragma unroll
        for (int mt = 0; mt < 4; ++mt) a[mt] = *(const v2f*)(Abuf + aBase[mt] + kk);
        #pragma unroll
        for (int nt = 0; nt < 2; ++nt) b[nt] = *(const v2f*)(Bbuf + bBase[nt] + kk);
        #pragma unroll
        for (int mt = 0; mt < 4; ++mt)
            #pragma unroll
            for (int nt = 0; nt < 2; ++nt)
                acc[mt][nt] = __builtin_amdgcn_wmma_f32_16x16x4_f32(
                    false, a[mt], false, b[nt], (short)0, acc[mt][nt], false, false);
    }
}

__global__ __launch_bounds__(256)
void gemm_wmma_f32(const float* __restrict__ L, const float* __restrict__ R,
                   float* __restrict__ sim) {
    __shared__ float As[2][128 * ASTR];
    __shared__ float Bs[2][128 * ASTR];

    const int tid  = threadIdx.x;
    const int lane = tid & 31;
    const int wv   = tid >> 5;           // 0..7
    const int wv_m = wv >> 2;            // 0..1  -> 64-row band
    const int wv_n = wv & 3;             // 0..3  -> 32-col band
    const int half = lane >> 4;          // K pair select {0,1} vs {2,3}
    const int lr   = lane & 15;
    const int ks   = half * 2;

    const int i0 = blockIdx.y * 128;     // L rows
    const int j0 = blockIdx.x * 128;     // R rows (sim cols)

    v8f acc[4][2];
    #pragma unroll
    for (int mt = 0; mt < 4; ++mt)
        #pragma unroll
        for (int nt = 0; nt < 2; ++nt)
            acc[mt][nt] = (v8f){0.f,0.f,0.f,0.f,0.f,0.f,0.f,0.f};

    int aBase[4], bBase[2];
    #pragma unroll
    for (int mt = 0; mt < 4; ++mt) aBase[mt] = (wv_m * 64 + mt * 16 + lr) * ASTR + ks;
    #pragma unroll
    for (int nt = 0; nt < 2; ++nt) bBase[nt] = (wv_n * 32 + nt * 16 + lr) * ASTR + ks;

    // per-thread staging slice: 4 float4 per matrix per chunk
    int srow[4], sc4[4];
    #pragma unroll
    for (int it = 0; it < 4; ++it) {
        int idx = tid + it * 256;        // 1024 float4 slots
        srow[it] = idx >> 3;             // 8 float4 per 32-wide row
        sc4[it]  = (idx & 7) << 2;
    }

#ifdef HAVE_ASYNC_LDS
    // ---- async double-buffered pipeline --------------------------------
    #pragma unroll
    for (int it = 0; it < 4; ++it) {     // issue chunk 0 -> buffer 0
        __builtin_amdgcn_global_load_async_to_lds_b128(
            (gv4p)(L + (size_t)(i0 + srow[it]) * DD + sc4[it]),
            (lv4p)(&As[0][srow[it] * ASTR + sc4[it]]), 0, 0);
        __builtin_amdgcn_global_load_async_to_lds_b128(
            (gv4p)(R + (size_t)(j0 + srow[it]) * DD + sc4[it]),
            (lv4p)(&Bs[0][srow[it] * ASTR + sc4[it]]), 0, 0);
    }
    int cur = 0;
    for (int c = 0; c < NCHUNK; ++c) {
        asm volatile("s_wait_asynccnt 0x0" ::: "memory");  // own slice of chunk c in LDS
        __syncthreads();                                   // everyone's slice in; prev buf free
        if (c + 1 < NCHUNK) {
            int kb = (c + 1) * KC;
            #pragma unroll
            for (int it = 0; it < 4; ++it) {
                __builtin_amdgcn_global_load_async_to_lds_b128(
                    (gv4p)(L + (size_t)(i0 + srow[it]) * DD + kb + sc4[it]),
                    (lv4p)(&As[cur ^ 1][srow[it] * ASTR + sc4[it]]), 0, 0);
                __builtin_amdgcn_global_load_async_to_lds_b128(
                    (gv4p)(R + (size_t)(j0 + srow[it]) * DD + kb + sc4[it]),
                    (lv4p)(&Bs[cur ^ 1][srow[it] * ASTR + sc4[it]]), 0, 0);
            }
        }
        compute_chunk(As[cur], Bs[cur], aBase, bBase, acc);
        cur ^= 1;
    }
#else
    // ---- fallback: register-prefetch double buffer ---------------------
    float4 pa[4], pb[4];
    #pragma unroll
    for (int it = 0; it < 4; ++it) {
        pa[it] = *(const float4*)(L + (size_t)(i0 + srow[it]) * DD + sc4[it]);
        pb[it] = *(const float4*)(R + (size_t)(j0 + srow[it]) * DD + sc4[it]);
    }
    #pragma unroll
    for (int it = 0; it < 4; ++it) {
        *(float4*)(&As[0][srow[it] * ASTR + sc4[it]]) = pa[it];
        *(float4*)(&Bs[0][srow[it] * ASTR + sc4[it]]) = pb[it];
    }
    int cur = 0;
    for (int c = 0; c < NCHUNK; ++c) {
        __syncthreads();                 // chunk c visible; prev buf fully consumed
        if (c + 1 < NCHUNK) {
            int kb = (c + 1) * KC;
            #pragma unroll
            for (int it = 0; it < 4; ++it) {
                pa[it] = *(const float4*)(L + (size_t)(i0 + srow[it]) * DD + kb + sc4[it]);
                pb[it] = *(const float4*)(R + (size_t)(j0 + srow[it]) * DD + kb + sc4[it]);
            }
        }
        compute_chunk(As[cur], Bs[cur], aBase, bBase, acc);
        if (c + 1 < NCHUNK) {
            #pragma unroll
            for (int it = 0; it < 4; ++it) {
                *(float4*)(&As[cur ^ 1][srow[it] * ASTR + sc4[it]]) = pa[it];
                *(float4*)(&Bs[cur ^ 1][srow[it] * ASTR + sc4[it]]) = pb[it];
            }
        }
        cur ^= 1;
    }
#endif

    // C/D layout: VGPR v holds row M = v + half*8, col N = lane&15
    #pragma unroll
    for (int mt = 0; mt < 4; ++mt)
        #pragma unroll
        for (int nt = 0; nt < 2; ++nt)
            #pragma unroll
            for (int v = 0; v < 8; ++v) {
                int row = i0 + wv_m * 64 + mt * 16 + half * 8 + v;
                int col = j0 + wv_n * 32 + nt * 16 + lr;
                sim[(size_t)row * NN + col] = acc[mt][nt][v];
            }
}

// ---------------------------------------------------------------------------
// Register-resident sorted top-10 (descending); constant indices after unroll
// ---------------------------------------------------------------------------
__device__ __forceinline__ void topk_insert(float (&t)[TK], float v) {
    if (v > t[TK - 1]) {
        t[TK - 1] = v;
        #pragma unroll
        for (int i = TK - 1; i > 0; --i) {
            if (t[i] > t[i - 1]) { float tmp = t[i - 1]; t[i - 1] = t[i]; t[i] = tmp; }
        }
    }
}

// Kernel 2: LR[r] = mean(top10 of sim row r). One wave32 per row.
__global__ __launch_bounds__(256)
void row_topk_mean(const float* __restrict__ sim, float* __restrict__ LR) {
    const int lane = threadIdx.x & 31;
    const int wv   = threadIdx.x >> 5;
    const int row  = blockIdx.x * 8 + wv;
    const float* p = sim + (size_t)row * NN;

    float t[TK];
    #pragma unroll
    for (int i = 0; i < TK; ++i) t[i] = -3.402823466e38f;

    for (int c = lane; c < NN; c += 32) topk_insert(t, p[c]);

    #pragma unroll
    for (int off = 16; off >= 1; off >>= 1) {
        float q[TK];
        #pragma unroll
        for (int i = 0; i < TK; ++i) q[i] = __shfl_xor(t[i], off, 32);
        #pragma unroll
        for (int i = 0; i < TK; ++i) topk_insert(t, q[i]);
    }

    if (lane == 0) {
        float s = 0.f;
        #pragma unroll
        for (int i = 0; i < TK; ++i) s += t[i];
        LR[row] = s * (1.0f / TK);
    }
}

// Kernel 3: RL[c] = mean(top10 of sim column c). Thread per column, coalesced.
__global__ __launch_bounds__(256)
void col_topk_mean(const float* __restrict__ sim, float* __restrict__ RL) {
    const int c = blockIdx.x * 256 + threadIdx.x;
    const float* p = sim + c;

    float t[TK];
    #pragma unroll
    for (int i = 0; i < TK; ++i) t[i] = -3.402823466e38f;

    #pragma unroll 4
    for (int r = 0; r < NN; ++r) topk_insert(t, p[(size_t)r * NN]);

    float s = 0.f;
    #pragma unroll
    for (int i = 0; i < TK; ++i) s += t[i];
    RL[c] = s * (1.0f / TK);
}

// Kernel 4: per csls-row r: rank of diagonal + row max.
// csls[r,l] = 2*sim[l,r] - LR[l] - RL[l]; stable-argsort tie break by index.
__global__ __launch_bounds__(256)
void rank_top1(const float* __restrict__ sim, const float* __restrict__ LR,
               const float* __restrict__ RL, int* __restrict__ ranks,
               float* __restrict__ top1) {
    const int r = blockIdx.x * 256 + threadIdx.x;
    const float diag = 2.0f * sim[(size_t)r * NN + r] - LR[r] - RL[r];
    const float* p = sim + r;

    float mx = -3.402823466e38f;
    int cnt_gt = 0, cnt_eq = 0;
    #pragma unroll 4
    for (int l = 0; l < NN; ++l) {
        float v = 2.0f * p[(size_t)l * NN] - LR[l] - RL[l];
        mx = fmaxf(mx, v);
        cnt_gt += (v > diag) ? 1 : 0;
        cnt_eq += ((v == diag) && (l < r)) ? 1 : 0;
    }
    ranks[r] = cnt_gt + cnt_eq;
    top1[r]  = mx;
}

// ---------------------------------------------------------------------------
extern "C" void kernel_launch(void* const* d_in, const int* in_sizes, int n_in,
                              void* d_out, int out_size, void* d_ws, size_t ws_size,
                              hipStream_t stream) {
    const float* Lvec = (const float*)d_in[0];
    const float* Rvec = (const float*)d_in[1];

    float* sim = (float*)d_ws;                       // 16384*16384 f32 = 1 GiB
    float* LR  = sim + (size_t)NN * NN;              // 16384 f32
    float* RL  = LR + NN;                            // 16384 f32

    int*   ranks = (int*)d_out;                      // first NN outputs (int32)
    float* top1  = (float*)d_out + NN;               // second NN outputs (f32)

    dim3 ggrid(NN / 128, NN / 128, 1);
    gemm_wmma_f32<<<ggrid, 256, 0, stream>>>(Lvec, Rvec, sim);

    row_topk_mean<<<NN / 8, 256, 0, stream>>>(sim, LR);
    col_topk_mean<<<NN / 256, 256, 0, stream>>>(sim, RL);
    rank_top1<<<NN / 256, 256, 0, stream>>>(sim, LR, RL, ranks, top1);
}